// ObservabilityWeightedMSE_81922206204659
// MI455X (gfx1250) — compile-verified
//
#include <hip/hip_runtime.h>
#include <stdint.h>

// ---------------------------------------------------------------------------
// ObservabilityWeightedMSE for MI455X (gfx1250)
// Pure streaming reduction: ~1.13 GB read, 1 float out -> HBM-bound (~48us
// floor @ 23.3 TB/s). wave32-per-row, ballot-based cutoff detection, and
// double-buffered CDNA5 async global->LDS staging of the 82-col pred/targ
// rows (prefetch next row while computing current; s_wait_asynccnt 4).
// ---------------------------------------------------------------------------

#define THRESH 1.0e-4f

#if defined(__HIP_DEVICE_COMPILE__)
#if __has_builtin(__builtin_amdgcn_global_load_async_to_lds_b64) && \
    __has_builtin(__builtin_amdgcn_s_wait_asynccnt)
#define DEV_ASYNC 1
#else
#define DEV_ASYNC 0
#endif
#else
#define DEV_ASYNC 0
#endif

typedef float v4f __attribute__((ext_vector_type(4)));
typedef int   v2i __attribute__((ext_vector_type(2)));

// spatial_mask[c] per the reference _build_spatial_mask()
__device__ __forceinline__ float spatial_w(int c) {
  if (c == 0 || c == 41) return 5.0f;
  if (c <= 40) return ((c & 3) == 0) ? 5.0f : 1.0f;  // 4,8,...,40
  int r = c - 44;                                     // 44,48,...,80 / 45,...,77
  if (r >= 0) {
    int q = r & 3;
    if (q == 0 && r <= 36) return 5.0f;
    if (q == 1 && r <= 33) return 5.0f;
  }
  return 1.0f;
}

#if DEV_ASYNC
__device__ __forceinline__ void g2l_b64(const float* g, float* l) {
  __builtin_amdgcn_global_load_async_to_lds_b64(
      (__attribute__((address_space(1))) v2i*)(g),
      (__attribute__((address_space(3))) v2i*)(l),
      /*offset=*/0, /*cpol=*/0);
}

// Issue 4 async b64 ops copying one 82-float pred row + targ row into the
// wave-private LDS buffers (rows are 328B apart -> always 8B-aligned).
__device__ __forceinline__ void prefetch_row(const float* pr, const float* tr,
                                             float* lp, float* lt, int lane) {
  g2l_b64(pr + 2 * lane, lp + 2 * lane);              // floats 0..63
  g2l_b64(tr + 2 * lane, lt + 2 * lane);
  if (lane < 9) {                                      // floats 64..81
    g2l_b64(pr + 64 + 2 * lane, lp + 64 + 2 * lane);
    g2l_b64(tr + 64 + 2 * lane, lt + 64 + 2 * lane);
  }
}
#endif

__global__ __launch_bounds__(256) void owmse_partial(
    const float* __restrict__ pred,
    const float* __restrict__ targ,
    const float* __restrict__ inp,
    float* __restrict__ partials,
    int B)
{
  __shared__ float ldsP[8][2][88];  // [wave][buffer][82 padded to 88]
  __shared__ float ldsT[8][2][88];
  __shared__ float wacc[8];

  const int lane = threadIdx.x & 31;
  const int wv   = threadIdx.x >> 5;
  const int gw   = blockIdx.x * (blockDim.x >> 5) + wv;
  const int nw   = gridDim.x * (blockDim.x >> 5);

  // Loop-invariant per-lane constants: columns lane, lane+32, lane+64.
  const int  c0 = lane, c1 = lane + 32, c2 = lane + 64;
  const bool has2 = (c2 < 82);                    // lanes 0..17
  const int  h0 = c0;                             // c % 41 positions
  const int  h1 = (c1 < 41) ? c1 : (c1 - 41);
  const int  h2 = c2 - 41;                        // c2 >= 64 > 41 always
  const float s0 = spatial_w(c0);
  const float s1 = spatial_w(c1);
  const float s2 = has2 ? spatial_w(c2) : 0.0f;

  float acc = 0.0f;
  int   buf = 0;

#if DEV_ASYNC
  if (gw < B) {
    prefetch_row(pred + (size_t)gw * 82, targ + (size_t)gw * 82,
                 ldsP[wv][0], ldsT[wv][0], lane);
  }
#endif

  for (int r = gw; r < B; r += nw) {
    // ---- stage-cutoff: lane j<10 owns node group j (20 floats at float
    //      offset j*40) = inputs_reshaped[:, j, 0, :]. 5x float4 NT loads.
    //      These overlap the in-flight async ops for this row's pred/targ.
    float m = -3.0e38f;
    if (lane < 10) {
      const v4f* gp = (const v4f*)(inp + (size_t)r * 400 + lane * 40);
#pragma unroll
      for (int i = 0; i < 5; ++i) {
        v4f v = __builtin_nontemporal_load(gp + i);
        m = fmaxf(m, fmaxf(fmaxf(v.x, v.y), fmaxf(v.z, v.w)));
      }
    }
    const unsigned long long bal = __ballot((lane < 10) && (m < THRESH));
    // first_idx = ctz(bal); cutoff = (first_idx+1)*4 = ffs(bal)*4; 1000 => none
    const int cutoff = bal ? (__ffsll(bal) * 4) : 1000;

    float p0, p1, p2, t0, t1, t2;
#if DEV_ASYNC
    {
      const int rn = r + nw;
      // DS reads of buf^1 (read in the previous iteration) must drain before async
      // writes can land there (async-LDS writes are unordered vs DS ops).
      asm volatile("s_wait_dscnt 0" ::: "memory");
      if (rn < B) {
        prefetch_row(pred + (size_t)rn * 82, targ + (size_t)rn * 82,
                     ldsP[wv][buf ^ 1], ldsT[wv][buf ^ 1], lane);
        // 8 async ops outstanding; retire the 4 belonging to current buf
        // (async loads complete in order).
        __builtin_amdgcn_s_wait_asynccnt(4);
      } else {
        __builtin_amdgcn_s_wait_asynccnt(0);
      }
      const float* lp = ldsP[wv][buf];
      const float* lt = ldsT[wv][buf];
      p0 = lp[c0]; p1 = lp[c1]; p2 = has2 ? lp[c2] : 0.0f;
      t0 = lt[c0]; t1 = lt[c1]; t2 = has2 ? lt[c2] : 0.0f;
      buf ^= 1;
    }
#else
    {
      const float* pr = pred + (size_t)r * 82;
      const float* tr = targ + (size_t)r * 82;
      p0 = pr[c0]; p1 = pr[c1]; p2 = has2 ? pr[c2] : 0.0f;
      t0 = tr[c0]; t1 = tr[c1]; t2 = has2 ? tr[c2] : 0.0f;
    }
#endif
    const float d0 = p0 - t0, d1 = p1 - t1, d2 = p2 - t2;
    const float w0 = (h0 >= cutoff) ? 0.1f : 1.0f;
    const float w1 = (h1 >= cutoff) ? 0.1f : 1.0f;
    const float w2 = (h2 >= cutoff) ? 0.1f : 1.0f;
    acc = fmaf(d0 * d0, s0 * w0, acc);
    acc = fmaf(d1 * d1, s1 * w1, acc);
    acc = fmaf(d2 * d2, s2 * w2, acc);
  }

  // One cross-lane reduction per wave at the very end (wave32).
#pragma unroll
  for (int off = 16; off > 0; off >>= 1)
    acc += __shfl_xor(acc, off, 32);
  if (lane == 0) wacc[wv] = acc;
  __syncthreads();
  if (threadIdx.x == 0) {
    float t = 0.0f;
    const int nwv = blockDim.x >> 5;
    for (int i = 0; i < nwv; ++i) t += wacc[i];
    partials[blockIdx.x] = t;   // every block writes -> d_ws fully owned
  }
}

__global__ __launch_bounds__(256) void owmse_final(
    const float* __restrict__ partials, int n, float scale,
    float* __restrict__ out)
{
  __shared__ float sm[256];
  float a = 0.0f;
  for (int i = threadIdx.x; i < n; i += blockDim.x) a += partials[i];
  sm[threadIdx.x] = a;
  __syncthreads();
  for (int o = 128; o > 0; o >>= 1) {
    if ((int)threadIdx.x < o) sm[threadIdx.x] += sm[threadIdx.x + o];
    __syncthreads();
  }
  if (threadIdx.x == 0) out[0] = sm[0] * scale;  // deterministic plain store
}

extern "C" void kernel_launch(void* const* d_in, const int* in_sizes, int n_in,
                              void* d_out, int out_size, void* d_ws, size_t ws_size,
                              hipStream_t stream)
{
  (void)n_in; (void)out_size;
  const float* pred = (const float*)d_in[0];
  const float* targ = (const float*)d_in[1];
  const float* inp  = (const float*)d_in[2];
  const int B = in_sizes[0] / 82;

  float* partials = (float*)d_ws;
  int blocks = 3072;                       // 24576 waves -> ~20 rows/wave
  size_t maxb = ws_size / sizeof(float);
  if ((size_t)blocks > maxb) blocks = (int)maxb;
  if (blocks < 1) blocks = 1;

  owmse_partial<<<blocks, 256, 0, stream>>>(pred, targ, inp, partials, B);
  const float scale = 1.0f / (82.0f * (float)B);
  owmse_final<<<1, 256, 0, stream>>>(partials, blocks, scale, (float*)d_out);
}